// PolarPointPillarFeatureNet_40235253629489
// MI455X (gfx1250) — compile-verified
//
#include <hip/hip_runtime.h>
#include <hip/hip_bf16.h>

// PolarPointPillar PFN for MI455X (gfx1250, wave32, WMMA).
//
//  k1:  stats pass: masked features -> E[f] (9) + E[f f^T] (45) partials
//       (wave permlane reduce -> LDS block reduce -> per-block slot, fixed order)
//  k1b: fixed-order sum of 256 block partials (deterministic, no atomics)
//  k2:  fold stats through W -> per-channel BN scale/shift
//  k3:  main: per pillar, K-permuted features -> 24x v_wmma_f32_16x16x4_f32
//       -> per-channel min/max over 32 points -> affine+relu -> out [N,64]
//       A fragments built with ZERO cross-lane ops (lane-split x,y / z,w loads);
//       all reductions use v_permlane16/v_permlanex16 (VALU, no LDS waits).

#define P 32
#define C_IN 4
#define C_OUT 64
#define C_FEAT 9
#define BN_EPS 1e-3f

#define STATS_BLOCKS 256
#define STATS_THREADS 256
#define NSTAT 54            // 9 first moments + 45 unique second moments
#define WS_SUMS 0           // ws[0..53]   final stat sums
#define WS_SCALE 64         // ws[64..127] per-channel scale
#define WS_SHIFT 128        // ws[128..191] per-channel shift
#define WS_PART 256         // ws[256..256+STATS_BLOCKS*54) block partials

typedef float v2f __attribute__((ext_vector_type(2)));
typedef float v8f __attribute__((ext_vector_type(8)));

// xor-butterfly lane exchange via v_permlane16_b32 (within row of 16) and
// v_permlanex16_b32 (across the two rows of a wave32). Pure VALU, no LDS.
template <int M>
__device__ __forceinline__ float fxor(float v) {
    int s = __float_as_int(v);
    int r;
    if constexpr (M == 1)
        r = __builtin_amdgcn_permlane16(s, s, 0x67452301u, 0xefcdab89u, false, false);
    else if constexpr (M == 2)
        r = __builtin_amdgcn_permlane16(s, s, 0x54761032u, 0xdcfe98bau, false, false);
    else if constexpr (M == 4)
        r = __builtin_amdgcn_permlane16(s, s, 0x32107654u, 0xba98fedcu, false, false);
    else if constexpr (M == 8)
        r = __builtin_amdgcn_permlane16(s, s, 0xfedcba98u, 0x76543210u, false, false);
    else  // M == 16: identity gather from the other row
        r = __builtin_amdgcn_permlanex16(s, s, 0x76543210u, 0xfedcba98u, false, false);
    return __int_as_float(r);
}

__device__ __forceinline__ float row_sum16(float v) {  // sum within each row of 16
    v += fxor<1>(v); v += fxor<2>(v); v += fxor<4>(v); v += fxor<8>(v);
    return v;
}
__device__ __forceinline__ float wave_sum32(float v) {
    v = row_sum16(v); v += fxor<16>(v);
    return v;
}

// Per-lane (lane == point) masked 9-feature vector (stats pass only).
__device__ __forceinline__ void compute_features(const float4* __restrict__ vox4,
                                                 const int* __restrict__ npp,
                                                 const float2* __restrict__ vc,
                                                 int pillar, int lane, float f[C_FEAT]) {
    float4 v = vox4[(size_t)pillar * P + lane];
    int npts = npp[pillar];
    float m = (lane < npts) ? 1.0f : 0.0f;
    float sx = wave_sum32(v.x * m);
    float sy = wave_sum32(v.y * m);
    float sz = wave_sum32(v.z * m);
    float inv = 1.0f / (float)npts;
    float2 c = vc[pillar];
    f[0] = v.x * m;  f[1] = v.y * m;  f[2] = v.z * m;  f[3] = v.w * m;
    f[4] = (v.x - sx * inv) * m;  f[5] = (v.y - sy * inv) * m;  f[6] = (v.z - sz * inv) * m;
    f[7] = (v.x - c.x) * m;       f[8] = (v.y - c.y) * m;
}

// ---- Stats pass ----
__global__ void pfn_stats(const float4* __restrict__ vox4, const int* __restrict__ npp,
                          const float2* __restrict__ vc, float* __restrict__ ws, int N) {
    __shared__ float part[STATS_THREADS / 32][NSTAT];
    const int lane = threadIdx.x & 31;
    const int waveIn = threadIdx.x >> 5;
    const int wave = blockIdx.x * (STATS_THREADS / 32) + waveIn;
    const int nwaves = gridDim.x * (STATS_THREADS / 32);

    float s1[C_FEAT];
    float s2[45];
#pragma unroll
    for (int i = 0; i < C_FEAT; ++i) s1[i] = 0.0f;
#pragma unroll
    for (int i = 0; i < 45; ++i) s2[i] = 0.0f;

    for (int pillar = wave; pillar < N; pillar += nwaves) {
        float f[C_FEAT];
        compute_features(vox4, npp, vc, pillar, lane, f);
        int idx = 0;
#pragma unroll
        for (int j = 0; j < C_FEAT; ++j) {
            s1[j] += f[j];
#pragma unroll
            for (int k = j; k < C_FEAT; ++k) s2[idx++] += f[j] * f[k];
        }
    }
#pragma unroll
    for (int i = 0; i < C_FEAT; ++i) {
        float v = wave_sum32(s1[i]);
        if (lane == 0) part[waveIn][i] = v;
    }
#pragma unroll
    for (int i = 0; i < 45; ++i) {
        float v = wave_sum32(s2[i]);
        if (lane == 0) part[waveIn][C_FEAT + i] = v;
    }
    __syncthreads();
    if (threadIdx.x < NSTAT) {
        float v = 0.0f;
#pragma unroll
        for (int w = 0; w < STATS_THREADS / 32; ++w) v += part[w][threadIdx.x];
        ws[WS_PART + (size_t)blockIdx.x * NSTAT + threadIdx.x] = v;
    }
}

__global__ void pfn_stats_final(float* __restrict__ ws) {
    int i = threadIdx.x;
    if (i >= NSTAT) return;
    float v = 0.0f;
    for (int b = 0; b < STATS_BLOCKS; ++b) v += ws[WS_PART + (size_t)b * NSTAT + i];
    ws[WS_SUMS + i] = v;
}

// ---- Fold stats through W: per-channel BN scale/shift ----
__global__ void pfn_finalize(const float* __restrict__ W, const float* __restrict__ gamma,
                             const float* __restrict__ beta, float* __restrict__ ws, float NP) {
    int o = threadIdx.x;
    if (o >= C_OUT) return;
    float w[C_FEAT];
#pragma unroll
    for (int c = 0; c < C_FEAT; ++c) w[c] = W[o * C_FEAT + c];
    float invNP = 1.0f / NP;
    float m = 0.0f;
#pragma unroll
    for (int c = 0; c < C_FEAT; ++c) m += w[c] * ws[WS_SUMS + c];
    m *= invNP;
    float e2 = 0.0f;
    int idx = 0;
#pragma unroll
    for (int j = 0; j < C_FEAT; ++j)
#pragma unroll
        for (int k = j; k < C_FEAT; ++k) {
            float v = ws[WS_SUMS + C_FEAT + idx++] * invNP;
            e2 += w[j] * w[k] * v * ((j == k) ? 1.0f : 2.0f);
        }
    float var = e2 - m * m;
    float scale = gamma[o] * rsqrtf(var + BN_EPS);
    ws[WS_SCALE + o] = scale;
    ws[WS_SHIFT + o] = beta[o] - m * scale;
}

// ---- Main pass: one wave per pillar, 24x v_wmma_f32_16x16x4_f32, zero LDS ----
//
// K-slot permutation (applied to both A and B, legal since WMMA sums over K):
//   low-lane slots  {0,1}:{x,y}  {4,5}:{x-mx, y-my}  {8,9}:{x-cx, y-cy}
//   high-lane slots {2,3}:{z,w}  {6,7}:{z-mz, 0}     {10,11}:{0, 0}
// Lanes 0-15 load the (x,y) halves of their two points; lanes 16-31 the (z,w)
// halves -> every lane builds its A elements locally, no cross-lane ops.
__global__ void __launch_bounds__(32)
pfn_main(const float2* __restrict__ voxf2, const int* __restrict__ npp,
         const float2* __restrict__ vc, const float* __restrict__ W,
         const float* __restrict__ ws, float* __restrict__ out, int N) {
    const int lane = threadIdx.x;
    const int col = lane & 15;
    const bool hi = lane >= 16;

    // B fragments (K-permuted W, 12x64), lanes: N=col; low regs K=4kc+{0,1},
    // high regs K=4kc+{2,3}.
    v2f B[3][4];
#pragma unroll
    for (int nt = 0; nt < 4; ++nt) {
        const float* wr = W + (nt * 16 + col) * C_FEAT;
        B[0][nt][0] = wr[hi ? 2 : 0];
        B[0][nt][1] = wr[hi ? 3 : 1];
        B[1][nt][0] = wr[hi ? 6 : 4];
        B[1][nt][1] = hi ? 0.0f : wr[5];
        B[2][nt][0] = hi ? 0.0f : wr[7];
        B[2][nt][1] = hi ? 0.0f : wr[8];
    }
    float sc[4], sh[4];
#pragma unroll
    for (int nt = 0; nt < 4; ++nt) {
        sc[nt] = ws[WS_SCALE + nt * 16 + col];
        sh[nt] = ws[WS_SHIFT + nt * 16 + col];
    }

    for (int pillar = blockIdx.x; pillar < N; pillar += gridDim.x) {
        const size_t pbase = (size_t)pillar * (P * 2);  // float2 units
        // prefetch next pillar's streaming data (speculative, OOB-safe)
        __builtin_prefetch(voxf2 + pbase + (size_t)gridDim.x * (P * 2) + col * 2 + (hi ? 1 : 0), 0, 0);

        int npts = npp[pillar];
        float2 c = vc[pillar];
        float2 d0 = voxf2[pbase + col * 2 + (hi ? 1 : 0)];            // pt = col
        float2 d1 = voxf2[pbase + (col + 16) * 2 + (hi ? 1 : 0)];     // pt = col+16
        float m0 = (col < npts) ? 1.0f : 0.0f;
        float m1 = (col + 16 < npts) ? 1.0f : 0.0f;

        // per-half sums: low row -> sum x (pa), sum y (pb); high row -> sum z (pa)
        float pa = row_sum16(d0.x * m0 + d1.x * m1);
        float pb = row_sum16(d0.y * m0 + d1.y * m1);
        float inv = 1.0f / (float)npts;
        float meanA = pa * inv;   // mx (low rows) / mz (high rows)
        float meanB = pb * inv;   // my (low rows) / unused (high rows)

        v8f Cacc[2][4] = {};
#pragma unroll
        for (int mt = 0; mt < 2; ++mt) {
            float2 d = mt ? d1 : d0;
            float m = mt ? m1 : m0;
            v2f A0, A1, A2;
            A0[0] = d.x * m;
            A0[1] = d.y * m;
            A1[0] = (d.x - meanA) * m;
            A1[1] = hi ? 0.0f : (d.y - meanB) * m;
            A2[0] = hi ? 0.0f : (d.x - c.x) * m;
            A2[1] = hi ? 0.0f : (d.y - c.y) * m;
#pragma unroll
            for (int nt = 0; nt < 4; ++nt)
                Cacc[mt][nt] = __builtin_amdgcn_wmma_f32_16x16x4_f32(
                    false, A0, false, B[0][nt], (short)0, Cacc[mt][nt], false, false);
#pragma unroll
            for (int nt = 0; nt < 4; ++nt)
                Cacc[mt][nt] = __builtin_amdgcn_wmma_f32_16x16x4_f32(
                    false, A1, false, B[1][nt], (short)0, Cacc[mt][nt], false, false);
#pragma unroll
            for (int nt = 0; nt < 4; ++nt)
                Cacc[mt][nt] = __builtin_amdgcn_wmma_f32_16x16x4_f32(
                    false, A2, false, B[2][nt], (short)0, Cacc[mt][nt], false, false);
        }
        // D layout: lane = channel col (rows 0-7 in low lanes, 8-15 in high lanes)
        float r[4];
#pragma unroll
        for (int nt = 0; nt < 4; ++nt) {
            float mx = Cacc[0][nt][0], mn = Cacc[0][nt][0];
#pragma unroll
            for (int q = 1; q < 8; ++q) { mx = fmaxf(mx, Cacc[0][nt][q]); mn = fminf(mn, Cacc[0][nt][q]); }
#pragma unroll
            for (int q = 0; q < 8; ++q) { mx = fmaxf(mx, Cacc[1][nt][q]); mn = fminf(mn, Cacc[1][nt][q]); }
            mx = fmaxf(mx, fxor<16>(mx));
            mn = fminf(mn, fxor<16>(mn));
            // relu(scale*x+shift) monotone in x iff scale>=0; else use min.
            float val = (sc[nt] >= 0.0f) ? mx : mn;
            r[nt] = fmaxf(0.0f, sc[nt] * val + sh[nt]);
        }
        // full-wave stores: low lanes write tiles 0,1 (ch 0-31), high lanes 2,3.
        size_t obase = (size_t)pillar * C_OUT + col + (hi ? 32 : 0);
        out[obase]      = hi ? r[2] : r[0];
        out[obase + 16] = hi ? r[3] : r[1];
    }
}

extern "C" void kernel_launch(void* const* d_in, const int* in_sizes, int n_in,
                              void* d_out, int out_size, void* d_ws, size_t ws_size,
                              hipStream_t stream) {
    const float* voxels = (const float*)d_in[0];
    const int* npp = (const int*)d_in[1];
    const float2* vc = (const float2*)d_in[2];
    const float* W = (const float*)d_in[3];
    const float* gamma = (const float*)d_in[4];
    const float* beta = (const float*)d_in[5];
    float* out = (float*)d_out;
    float* ws = (float*)d_ws;

    int N = in_sizes[0] / (P * C_IN);
    float NP = (float)N * (float)P;

    pfn_stats<<<STATS_BLOCKS, STATS_THREADS, 0, stream>>>(
        (const float4*)voxels, npp, vc, ws, N);
    pfn_stats_final<<<1, 64, 0, stream>>>(ws);
    pfn_finalize<<<1, 64, 0, stream>>>(W, gamma, beta, ws, NP);

    pfn_main<<<16384, 32, 0, stream>>>(
        (const float2*)voxels, npp, vc, W, ws, out, N);
}